// CombineGraph_51359218925876
// MI455X (gfx1250) — compile-verified
//
#include <hip/hip_runtime.h>
#include <hip/hip_bf16.h>

#define BB 128
#define SS 50
#define DD 128
#define NNODE 40000
#define NSAMP 12
#define NNEI 8
#define FTEMP 0.5f
#define FMU 0.1f
#define FLEAK 0.2f

typedef __attribute__((ext_vector_type(2))) float v2f;
typedef __attribute__((ext_vector_type(8))) float v8f;

// ---------------------------------------------------------------------------
// WMMA f32 16x16x4 helpers (full fp32 precision; matches reference numerics)
// ---------------------------------------------------------------------------
__device__ __forceinline__ v8f wmma4(v2f a, v2f b, v8f c) {
  return __builtin_amdgcn_wmma_f32_16x16x4_f32(false, a, false, b, (short)0, c,
                                               false, false);
}

// A-side fragment: row-major matrix, frag rows (M or N of A@B^T) mapped to
// lanes, K along the two regs.  lanes 0-15: row r0+lane, K=k,k+1;
// lanes 16-31: row r0+lane-16, K=k+2,k+3.  kk is always even and every call
// site has 8-byte-aligned rows -> single b64 load.
__device__ __forceinline__ v2f frag_rows(const float* M, int ld, int r0, int k) {
  int lane = threadIdx.x & 31;
  long r = r0 + (lane & 15);
  int kk = k + ((lane >> 4) << 1);
  return *(const v2f*)(M + r * (long)ld + kk);
}

__device__ __forceinline__ v2f frag_rows_g(const float* M, int ld, int r0, int k,
                                           int rmax, int kmax) {
  int lane = threadIdx.x & 31;
  long r = r0 + (lane & 15);
  int kk = k + ((lane >> 4) << 1);
  v2f f;
  f.x = 0.f; f.y = 0.f;
  if (r < rmax) {
    if (kk + 1 < kmax)  f = *(const v2f*)(M + r * (long)ld + kk);
    else if (kk < kmax) f.x = M[r * (long)ld + kk];
  }
  return f;
}

// B-side fragment for W (K x N, row-major): lanes map to N, K along regs
// (strided by ld -> stays scalar).
__device__ __forceinline__ v2f frag_w(const float* W, int ld, int n0, int k) {
  int lane = threadIdx.x & 31;
  int n = n0 + (lane & 15);
  int kk = k + ((lane >> 4) << 1);
  v2f f;
  f.x = W[(long)kk * ld + n];
  f.y = W[(long)(kk + 1) * ld + n];
  return f;
}

__device__ __forceinline__ v2f frag_w_g(const float* W, int ld, int n0, int k,
                                        int kmax) {
  int lane = threadIdx.x & 31;
  int n = n0 + (lane & 15);
  int kk = k + ((lane >> 4) << 1);
  v2f f;
  f.x = (kk < kmax)     ? W[(long)kk * ld + n]       : 0.f;
  f.y = (kk + 1 < kmax) ? W[(long)(kk + 1) * ld + n] : 0.f;
  return f;
}

__device__ __forceinline__ float sigmoidf(float v) { return 1.f / (1.f + expf(-v)); }
__device__ __forceinline__ float leakyf(float v) { return v > 0.f ? v : FLEAK * v; }

// ---------------------------------------------------------------------------
// Generic WMMA GEMM: O(M x 128) = epi( A1(M x 128)@W1(128x128) [+ A2@W2] )
// grid.x = M/16, block = 256 (8 waves, wave w -> columns 16w..16w+15)
// K fixed at 128 -> fully unrollable WMMA pipeline.
// EPI: 0 plain, 1 tanh, 2 gate(P,Q), 3 mirror gate, 4 gate*MU, 5 sigmoid+row-add,
//      6 +bias
// ---------------------------------------------------------------------------
template <int EPI>
__global__ __launch_bounds__(256) void gemm128_kernel(
    const float* A1, int lda1, const float* A2, int lda2, const float* W1,
    const float* W2, const float* P, const float* Q, const float* bias,
    const float* addrow, int rows_per_b, float* O1, float* O2, int M) {
  int m0 = blockIdx.x * 16;
  int w = threadIdx.x >> 5;
  int n0 = w * 16;
  v8f acc = {};
#pragma unroll 8
  for (int k = 0; k < 128; k += 4) {
    v2f a = frag_rows(A1, lda1, m0, k);
    v2f b = frag_w(W1, 128, n0, k);
    acc = wmma4(a, b, acc);
  }
  if (A2 != nullptr) {
#pragma unroll 8
    for (int k = 0; k < 128; k += 4) {
      v2f a = frag_rows(A2, lda2, m0, k);
      v2f b = frag_w(W2, 128, n0, k);
      acc = wmma4(a, b, acc);
    }
  }
  __syncthreads();  // all waves done reading A -> safe in-place O1 == A1
  int lane = threadIdx.x & 31;
#pragma unroll
  for (int i = 0; i < 8; ++i) {
    int m = m0 + i + ((lane >> 4) << 3);
    int n = n0 + (lane & 15);
    float v = acc[i];
    long o = (long)m * 128 + n;
    if (EPI == 0) {
      O1[o] = v;
    } else if (EPI == 1) {
      O1[o] = tanhf(v);
    } else if (EPI == 2) {
      float g = sigmoidf(v);
      O1[o] = g * P[o] + (1.f - g) * Q[o];
    } else if (EPI == 3) {
      float g = sigmoidf(v);
      float p = P[o], q = Q[o];
      O1[o] = g * p + (1.f - g) * q;
      O2[o] = g * q + (1.f - g) * p;
    } else if (EPI == 4) {
      float g = sigmoidf(v) * FMU;
      O1[o] = g * P[o] + (1.f - g) * Q[o];
    } else if (EPI == 5) {
      int bidx = m / rows_per_b;
      O1[o] = sigmoidf(v + addrow[(long)bidx * 128 + n]);
    } else if (EPI == 6) {
      O1[o] = v + bias[n];
    }
  }
}

// ---------------------------------------------------------------------------
// Gathers: h = emb[inputs], hf1/hf2 = masked-mean pools over neighbor lists
// ---------------------------------------------------------------------------
__global__ __launch_bounds__(128) void gather_kernel(
    const int* inputs, const int* as0, const int* as1, const int* ssl0,
    const int* ssl1, const float* emb, float* h, float* hf1, float* hf2) {
  long row = blockIdx.x;  // B*S
  int d = threadIdx.x;
  int node = inputs[row];
  h[row * DD + d] = emb[(long)node * DD + d];
  const int* lists[4] = {as0, as1, ssl0, ssl1};
  float pv[4];
#pragma unroll
  for (int l = 0; l < 4; ++l) {
    float sum = 0.f, cnt = 0.f;
    for (int j = 0; j < NNEI; ++j) {
      int idx = lists[l][row * NNEI + j];
      if (idx != 0) {
        sum += emb[(long)idx * DD + d];
        cnt += 1.f;
      }
    }
    pv[l] = sum / (cnt + 1e-8f);
  }
  hf1[row * DD + d] = 0.5f * (pv[0] + pv[1]);
  hf2[row * DD + d] = 0.5f * (pv[2] + pv[3]);
}

__global__ __launch_bounds__(128) void sess_kernel(const int* inputs,
                                                   const int* item,
                                                   const float* emb,
                                                   float* sess) {
  int b = blockIdx.x;
  int d = threadIdx.x;
  float s = 0.f, c = 0.f;
  for (int t = 0; t < SS; ++t) {
    float mi = (inputs[b * SS + t] != 0) ? 1.f : 0.f;
    s += emb[(long)item[b * SS + t] * DD + d] * mi;
    c += mi;
  }
  sess[(long)b * DD + d] = s / c;
}

__global__ __launch_bounds__(128) void neigh1_kernel(const int* inputs,
                                                     const int* adj_all,
                                                     const float* emb,
                                                     int* neigh1, float* ev1) {
  long r = blockIdx.x;  // B*S*NSAMP
  int d = threadIdx.x;
  long prow = r / NSAMP;
  int j = (int)(r % NSAMP);
  int nid = adj_all[(long)inputs[prow] * NSAMP + j];
  if (d == 0) neigh1[r] = nid;
  ev1[r * DD + d] = emb[(long)nid * DD + d];
}

// ---------------------------------------------------------------------------
// Fused hop attention: for one sv row, gather 12 neighbors, compute
// a = leaky((sess*nv | w) @ W1)  (16x132 @ 132x128 via WMMA, 8 waves),
// logits = a @ W2, softmax over 12, agg = sum alpha_j * nv_j.
// mode 0: neighbors are contiguous rows of nv_contig;  mode 1: emb gather
// (with a prefetch of all 12 rows to warm L2 while LDS is being zeroed).
// ---------------------------------------------------------------------------
__global__ __launch_bounds__(256) void hop_attn_kernel(
    const int* parent, const int* adj_all, const float* num, const float* emb,
    const float* nv_contig, const float* sess, const float* W1,
    const float* W2, float* agg, int rows_per_b, int mode) {
  long r = blockIdx.x;
  int t = threadIdx.x;
  int b = (int)(r / rows_per_b);
  __shared__ float nv_s[NSAMP][DD];
  __shared__ float snv_s[16][132];
  __shared__ float a_s[16][DD];
  __shared__ float alpha_s[16];
  __shared__ float logit_s[16];
  int p = parent[r];
  if (mode == 1 && t < NSAMP) {  // warm the strided emb rows early
    int nid = adj_all[(long)p * NSAMP + t];
    __builtin_prefetch(emb + (long)nid * DD, 0, 1);
  }
  for (int i = t; i < 16 * 132; i += 256) ((float*)snv_s)[i] = 0.f;
  __syncthreads();
  for (int i = t; i < NSAMP * DD; i += 256) {
    int j = i / DD, d = i % DD;
    float v;
    if (mode == 0) {
      v = nv_contig[((long)r * NSAMP + j) * DD + d];
    } else {
      int nid = adj_all[(long)p * NSAMP + j];
      v = emb[(long)nid * DD + d];
    }
    nv_s[j][d] = v;
    snv_s[j][d] = v * sess[(long)b * DD + d];
  }
  if (t < NSAMP) snv_s[t][128] = num[(long)p * NSAMP + t];
  __syncthreads();

  int w = t >> 5, lane = t & 31;
  int n0 = w * 16;
  int koff = (lane >> 4) << 1;
  v8f acc = {};
#pragma unroll
  for (int k = 0; k < 132; k += 4) {
    int kk = k + koff;
    v2f a;
    a.x = snv_s[lane & 15][kk];
    a.y = snv_s[lane & 15][kk + 1];
    v2f bf;
    bf.x = (kk < 129) ? W1[(long)kk * 128 + n0 + (lane & 15)] : 0.f;
    bf.y = (kk + 1 < 129) ? W1[(long)(kk + 1) * 128 + n0 + (lane & 15)] : 0.f;
    acc = wmma4(a, bf, acc);
  }
#pragma unroll
  for (int i = 0; i < 8; ++i) {
    int m = i + ((lane >> 4) << 3);
    int n = n0 + (lane & 15);
    a_s[m][n] = leakyf(acc[i]);
  }
  __syncthreads();
  if (t < 16) {  // deterministic per-row dot for the logits
    float s = 0.f;
    for (int n = 0; n < DD; ++n) s += a_s[t][n] * W2[n];
    logit_s[t] = s;
  }
  __syncthreads();
  if (t == 0) {
    float mx = -3.4e38f;
    for (int j = 0; j < NSAMP; ++j) mx = fmaxf(mx, logit_s[j]);
    float sum = 0.f;
    for (int j = 0; j < NSAMP; ++j) {
      float e = expf(logit_s[j] - mx);
      alpha_s[j] = e;
      sum += e;
    }
    for (int j = 0; j < NSAMP; ++j) alpha_s[j] /= sum;
  }
  __syncthreads();
  if (t < DD) {
    float s = 0.f;
    for (int j = 0; j < NSAMP; ++j) s += alpha_s[j] * nv_s[j][t];
    agg[r * (long)DD + t] = s;
  }
}

// ---------------------------------------------------------------------------
// Local graph attention: e[k,b,i,j] = sum_d x[b,i,d]*aL[k,d]*x[b,j,d]
// (4 scaled-A WMMA accumulators per tile), then att = where(adj==k+1, e, -9e15)
// ---------------------------------------------------------------------------
__global__ __launch_bounds__(32) void local_att_kernel(const float* x,
                                                       const float* aL,
                                                       const int* adj,
                                                       float* att) {
  int bb = blockIdx.z;
  int m0 = blockIdx.y * 16;
  int n0 = blockIdx.x * 16;
  const float* X = x + (long)bb * SS * DD;
  int lane = threadIdx.x & 31;
  int koff = (lane >> 4) << 1;
  v8f e0 = {}, e1 = {}, e2 = {}, e3 = {};
#pragma unroll 4
  for (int k = 0; k < DD; k += 4) {
    v2f a = frag_rows_g(X, DD, m0, k, SS, DD);
    v2f b = frag_rows_g(X, DD, n0, k, SS, DD);
    int kk = k + koff;
    v2f a0, a1, a2, a3;
    a0.x = a.x * aL[0 * DD + kk]; a0.y = a.y * aL[0 * DD + kk + 1];
    a1.x = a.x * aL[1 * DD + kk]; a1.y = a.y * aL[1 * DD + kk + 1];
    a2.x = a.x * aL[2 * DD + kk]; a2.y = a.y * aL[2 * DD + kk + 1];
    a3.x = a.x * aL[3 * DD + kk]; a3.y = a.y * aL[3 * DD + kk + 1];
    e0 = wmma4(a0, b, e0);
    e1 = wmma4(a1, b, e1);
    e2 = wmma4(a2, b, e2);
    e3 = wmma4(a3, b, e3);
  }
#pragma unroll
  for (int i = 0; i < 8; ++i) {
    int m = m0 + i + ((lane >> 4) << 3);
    int n = n0 + (lane & 15);
    if (m < SS && n < SS) {
      int av = adj[((long)bb * SS + m) * SS + n];
      float v = -9e15f;
      if (av == 1) v = leakyf(e0[i]);
      else if (av == 2) v = leakyf(e1[i]);
      else if (av == 3) v = leakyf(e2[i]);
      else if (av == 4) v = leakyf(e3[i]);
      att[((long)bb * SS + m) * SS + n] = v;
    }
  }
}

__global__ __launch_bounds__(64) void softmax50_kernel(float* att) {
  long row = blockIdx.x;  // B*S rows of length 50
  float* p = att + row * SS;
  __shared__ float red[64];
  int t = threadIdx.x;
  float v = (t < SS) ? p[t] : -3.4e38f;
  red[t] = v;
  __syncthreads();
  for (int o = 32; o > 0; o >>= 1) {
    if (t < o) red[t] = fmaxf(red[t], red[t + o]);
    __syncthreads();
  }
  float mx = red[0];
  __syncthreads();
  float e = (t < SS) ? expf(v - mx) : 0.f;
  red[t] = e;
  __syncthreads();
  for (int o = 32; o > 0; o >>= 1) {
    if (t < o) red[t] += red[t + o];
    __syncthreads();
  }
  if (t < SS) p[t] = e / red[0];
}

// out(50x128) = attn_b(50x50) @ x_b(50x128), batched over b
__global__ __launch_bounds__(32) void attn_x_kernel(const float* attn,
                                                    const float* x,
                                                    float* out) {
  int bb = blockIdx.z;
  int m0 = blockIdx.y * 16;
  int n0 = blockIdx.x * 16;
  const float* A = attn + (long)bb * SS * SS;
  const float* W = x + (long)bb * SS * DD;
  v8f acc = {};
#pragma unroll
  for (int k = 0; k < SS; k += 4) {
    v2f a = frag_rows_g(A, SS, m0, k, SS, SS);
    v2f b = frag_w_g(W, DD, n0, k, SS);
    acc = wmma4(a, b, acc);
  }
  int lane = threadIdx.x & 31;
#pragma unroll
  for (int i = 0; i < 8; ++i) {
    int m = m0 + i + ((lane >> 4) << 3);
    int n = n0 + (lane & 15);
    if (m < SS) out[((long)bb * SS + m) * DD + n] = acc[i];
  }
}

__global__ __launch_bounds__(128) void hlocal_kernel(const float* xdot,
                                                     const int* lm,
                                                     float* hlocal) {
  int b = blockIdx.x;
  int d = threadIdx.x;
  float s = 0.f;
  for (int t = 0; t < SS; ++t)
    if (lm[b * SS + t] != 0) s += xdot[((long)b * SS + t) * DD + d];
  hlocal[(long)b * DD + d] = s;
}

__global__ __launch_bounds__(128) void hs_kernel(const float* hg,
                                                 const int* inputs, float* hs) {
  int b = blockIdx.x;
  int d = threadIdx.x;
  float s = 0.f, c = 0.f;
  for (int t = 0; t < SS; ++t) {
    float mi = (inputs[b * SS + t] != 0) ? 1.f : 0.f;
    s += hg[((long)b * SS + t) * DD + d] * mi;
    c += mi;
  }
  hs[(long)b * DD + d] = s / c;
}

__global__ __launch_bounds__(128) void hp_kernel(const float* xdot,
                                                 const float* pos, float* hp) {
  long row = blockIdx.x;
  int d = threadIdx.x;
  int s = (int)(row % SS);
  hp[row * DD + d] = xdot[row * DD + d] + pos[(long)s * DD + d];
}

__global__ __launch_bounds__(128) void betazg_kernel(const float* nh,
                                                     const float* hp,
                                                     const float* w_s,
                                                     const int* inputs,
                                                     float* zg) {
  int b = blockIdx.x;
  int t = threadIdx.x;
  __shared__ float beta[SS];
  if (t < SS) {
    const float* row = nh + ((long)b * SS + t) * DD;
    float s = 0.f;
    for (int d = 0; d < DD; ++d) s += row[d] * w_s[d];
    float mi = (inputs[b * SS + t] != 0) ? 1.f : 0.f;
    beta[t] = s * mi;
  }
  __syncthreads();
  float z = 0.f;
  for (int s2 = 0; s2 < SS; ++s2) z += beta[s2] * hp[((long)b * SS + s2) * DD + t];
  zg[(long)b * DD + t] = z;
}

// sim[b] = hf1_b @ hf2_b^T / TEMP  (batched WMMA A@B^T)
__global__ __launch_bounds__(32) void sim_kernel(const float* hf1,
                                                 const float* hf2, float* sim) {
  int bb = blockIdx.z;
  int m0 = blockIdx.y * 16;
  int n0 = blockIdx.x * 16;
  const float* A = hf1 + (long)bb * SS * DD;
  const float* Bm = hf2 + (long)bb * SS * DD;
  v8f acc = {};
#pragma unroll 4
  for (int k = 0; k < DD; k += 4) {
    v2f a = frag_rows_g(A, DD, m0, k, SS, DD);
    v2f b = frag_rows_g(Bm, DD, n0, k, SS, DD);
    acc = wmma4(a, b, acc);
  }
  int lane = threadIdx.x & 31;
#pragma unroll
  for (int i = 0; i < 8; ++i) {
    int m = m0 + i + ((lane >> 4) << 3);
    int n = n0 + (lane & 15);
    if (m < SS && n < SS)
      sim[((long)bb * SS + m) * SS + n] = acc[i] * (1.f / FTEMP);
  }
}

__global__ __launch_bounds__(64) void loss_kernel(const float* sim,
                                                  const int* smask,
                                                  float* partial) {
  int b = blockIdx.x;
  int t = threadIdx.x;
  float rs = 0.f;
  if (t < SS) {
    const float* row = sim + ((long)b * SS + t) * SS;
    const int* mrow = smask + ((long)b * SS + t) * SS;
    float mx = -3.4e38f;
    for (int j = 0; j < SS; ++j) mx = fmaxf(mx, row[j]);
    float sum = 0.f;
    for (int j = 0; j < SS; ++j) sum += expf(row[j] - mx);
    for (int j = 0; j < SS; ++j) {
      float sm = expf(row[j] - mx) / sum;
      if (mrow[j] == 1) rs += -logf(sm + 1e-8f);
    }
  }
  __shared__ float red[64];
  red[t] = rs;
  __syncthreads();
  for (int o = 32; o > 0; o >>= 1) {
    if (t < o) red[t] += red[t + o];
    __syncthreads();
  }
  if (t == 0) partial[b] = red[0];
}

__global__ void loss_final_kernel(const float* partial, float* out0) {
  if (threadIdx.x == 0 && blockIdx.x == 0) {
    float s = 0.f;
    for (int b = 0; b < BB; ++b) s += partial[b];
    out0[0] = s / (float)BB;
  }
}

// scores = zh(128x128) @ emb[1:]^T  -> (128, 39999)
__global__ __launch_bounds__(32) void scores_kernel(const float* zh,
                                                    const float* emb,
                                                    float* out) {
  const int NV = NNODE - 1;
  int n0 = blockIdx.x * 16;
  int m0 = blockIdx.y * 16;
  const float* bw = emb + DD;  // emb[1:]
  v8f acc = {};
#pragma unroll 8
  for (int k = 0; k < DD; k += 4) {
    v2f a = frag_rows(zh, DD, m0, k);
    v2f b = frag_rows_g(bw, DD, n0, k, NV, DD);
    acc = wmma4(a, b, acc);
  }
  int lane = threadIdx.x & 31;
#pragma unroll
  for (int i = 0; i < 8; ++i) {
    int m = m0 + i + ((lane >> 4) << 3);
    int n = n0 + (lane & 15);
    if (n < NV) out[(long)m * NV + n] = acc[i];
  }
}

// ---------------------------------------------------------------------------
extern "C" void kernel_launch(void* const* d_in, const int* in_sizes, int n_in,
                              void* d_out, int out_size, void* d_ws,
                              size_t ws_size, hipStream_t stream) {
  (void)in_sizes; (void)n_in; (void)out_size; (void)ws_size;
  const int* inputs    = (const int*)d_in[0];
  const int* adj       = (const int*)d_in[1];
  const int* item      = (const int*)d_in[2];
  const int* simi_mask = (const int*)d_in[3];
  const int* as0       = (const int*)d_in[4];
  const int* as1       = (const int*)d_in[5];
  const int* ssl0      = (const int*)d_in[6];
  const int* ssl1      = (const int*)d_in[7];
  const int* lm        = (const int*)d_in[8];
  const int* adj_all   = (const int*)d_in[9];
  const float* num     = (const float*)d_in[10];
  const float* emb     = (const float*)d_in[11];
  const float* pos     = (const float*)d_in[12];
  const float* a_local = (const float*)d_in[13];
  const float* mir_w1  = (const float*)d_in[14];
  const float* mir_w2  = (const float*)d_in[15];
  const float* glob_w1 = (const float*)d_in[16];
  const float* glob_w2 = (const float*)d_in[17];
  const float* glob_w3 = (const float*)d_in[18];
  const float* attr_w  = (const float*)d_in[19];
  const float* highway_w = (const float*)d_in[20];
  const float* glu1_w  = (const float*)d_in[21];
  const float* glu2_w  = (const float*)d_in[22];
  const float* glu4_w  = (const float*)d_in[23];
  const float* glu4_b  = (const float*)d_in[24];
  const float* w_s     = (const float*)d_in[25];
  const float* gate_w  = (const float*)d_in[26];
  float* out = (float*)d_out;

  // ---- workspace carve (floats) ----
  const long BSD = (long)BB * SS * DD;          // 819200
  const long R1D = (long)BB * SS * NSAMP * DD;  // 9830400
  float* ws = (float*)d_ws;
  long off = 0;
  auto alloc = [&](long n) { float* p = ws + off; off += n; return p; };
  float* h    = alloc(BSD);
  float* hf1  = alloc(BSD);
  float* hf2  = alloc(BSD);
  float* hf   = alloc(BSD);
  float* x2   = alloc(BSD);
  float* xA   = alloc(BSD);
  float* mA   = alloc(BSD);
  float* xB   = alloc(BSD);
  float* mB   = alloc(BSD);
  float* xdot = alloc(BSD);
  float* hpb  = alloc(BSD);
  float* nhb  = alloc(BSD);
  float* ev0  = alloc(BSD);
  float* ev1  = alloc(R1D);
  float* agg  = alloc(R1D);
  float* attb = alloc((long)BB * SS * SS);
  float* simb = alloc((long)BB * SS * SS);
  float* sess = alloc((long)BB * DD);
  float* hsb  = alloc((long)BB * DD);
  float* hloc = alloc((long)BB * DD);
  float* t2b  = alloc((long)BB * DD);
  float* zgb  = alloc((long)BB * DD);
  float* zhb  = alloc((long)BB * DD);
  float* lpar = alloc(BB);
  int* neigh1 = (int*)alloc((long)BB * SS * NSAMP);

  const int MBS = BB * SS;          // 6400 rows
  const int MR1 = BB * SS * NSAMP;  // 76800 rows
  const long DSQ = (long)DD * DD;   // 16384

  // 1) gathers + pools + session vector
  gather_kernel<<<MBS, 128, 0, stream>>>(inputs, as0, as1, ssl0, ssl1, emb, h,
                                         hf1, hf2);
  sess_kernel<<<BB, 128, 0, stream>>>(inputs, item, emb, sess);

  // 2) attribute gate: hf = sigma([h|hf1]@attr_w) * h + (1-..)*hf1
  gemm128_kernel<2><<<MBS / 16, 256, 0, stream>>>(
      h, DD, hf1, DD, attr_w, attr_w + 128 * 128, h, hf1, nullptr, nullptr, 1,
      hf, nullptr, MBS);

  // 3) global-hop branch
  neigh1_kernel<<<MR1, 128, 0, stream>>>(inputs, adj_all, emb, neigh1, ev1);
  hipMemcpyAsync(ev0, h, (size_t)BSD * sizeof(float), hipMemcpyDeviceToDevice,
                 stream);
  // n_hop=0, hop=0
  hop_attn_kernel<<<MBS, 256, 0, stream>>>(inputs, adj_all, num, emb, ev1,
                                           sess, glob_w1, glob_w2, agg, SS, 0);
  gemm128_kernel<1><<<MBS / 16, 256, 0, stream>>>(
      ev0, DD, agg, DD, glob_w3, glob_w3 + 128 * 128, nullptr, nullptr,
      nullptr, nullptr, 1, ev0, nullptr, MBS);
  // n_hop=0, hop=1 (neighbors gathered straight from emb -> no ev2 buffer)
  hop_attn_kernel<<<MR1, 256, 0, stream>>>(neigh1, adj_all, num, emb, nullptr,
                                           sess, glob_w1, glob_w2, agg,
                                           SS * NSAMP, 1);
  gemm128_kernel<1><<<MR1 / 16, 256, 0, stream>>>(
      ev1, DD, agg, DD, glob_w3, glob_w3 + 128 * 128, nullptr, nullptr,
      nullptr, nullptr, 1, ev1, nullptr, MR1);
  // n_hop=1, hop=0
  hop_attn_kernel<<<MBS, 256, 0, stream>>>(inputs, adj_all, num, emb, ev1,
                                           sess, glob_w1 + 129 * 128,
                                           glob_w2 + 128, agg, SS, 0);
  gemm128_kernel<1><<<MBS / 16, 256, 0, stream>>>(
      ev0, DD, agg, DD, glob_w3 + 256 * 128,
      glob_w3 + 256 * 128 + 128 * 128, nullptr, nullptr, nullptr, nullptr, 1,
      ev0, nullptr, MBS);
  // ev0 is now h_global (B,S,D)

  // 4) local branch: 2 iterations of graph attention + mirror gating
  const float* xcur = h;
  const float* mcur = hf;
  float* xo[2] = {xA, xB};
  float* mo[2] = {mA, mB};
  for (int it = 0; it < 2; ++it) {
    local_att_kernel<<<dim3(4, 4, BB), 32, 0, stream>>>(
        xcur, a_local + (long)it * 4 * DD, adj, attb);
    softmax50_kernel<<<MBS, 64, 0, stream>>>(attb);
    attn_x_kernel<<<dim3(8, 4, BB), 32, 0, stream>>>(attb, xcur, x2);
    gemm128_kernel<3><<<MBS / 16, 256, 0, stream>>>(
        x2, DD, mcur, DD, mir_w1 + it * DSQ, mir_w2 + it * DSQ, x2, mcur,
        nullptr, nullptr, 1, xo[it], mo[it], MBS);
    xcur = xo[it];
    mcur = mo[it];
  }
  // highway: x_dot = g*h + (1-g)*x
  gemm128_kernel<2><<<MBS / 16, 256, 0, stream>>>(
      h, DD, xcur, DD, highway_w, highway_w + 128 * 128, h, xcur, nullptr,
      nullptr, 1, xdot, nullptr, MBS);
  hlocal_kernel<<<BB, 128, 0, stream>>>(xdot, lm, hloc);
  hs_kernel<<<BB, 128, 0, stream>>>(ev0, inputs, hsb);

  // 5) SSL similarity loss
  sim_kernel<<<dim3(4, 4, BB), 32, 0, stream>>>(hf1, hf2, simb);
  loss_kernel<<<BB, 64, 0, stream>>>(simb, simi_mask, lpar);
  loss_final_kernel<<<1, 32, 0, stream>>>(lpar, out);

  // 6) GLU readout
  hp_kernel<<<MBS, 128, 0, stream>>>(xdot, pos, hpb);
  gemm128_kernel<6><<<BB / 16, 256, 0, stream>>>(
      hsb, DD, hloc, DD, glu2_w, glu4_w, nullptr, nullptr, glu4_b, nullptr, 1,
      t2b, nullptr, BB);
  gemm128_kernel<5><<<MBS / 16, 256, 0, stream>>>(
      hpb, DD, nullptr, 0, glu1_w, nullptr, nullptr, nullptr, nullptr, t2b,
      SS, nhb, nullptr, MBS);
  betazg_kernel<<<BB, 128, 0, stream>>>(nhb, hpb, w_s, inputs, zgb);
  gemm128_kernel<4><<<BB / 16, 256, 0, stream>>>(
      zgb, DD, hloc, DD, gate_w, gate_w + 128 * 128, hloc, zgb, nullptr,
      nullptr, 1, zhb, nullptr, BB);

  // 7) scores = zh @ emb[1:]^T
  scores_kernel<<<dim3((NNODE - 1 + 15) / 16, BB / 16), 32, 0, stream>>>(
      zhb, emb, out + 1);
}